// encoder_26920855011669
// MI455X (gfx1250) — compile-verified
//
#include <hip/hip_runtime.h>
#include <hip/hip_bf16.h>
#include <cstdint>

// ---------------------------------------------------------------------------
// MI455X / gfx1250 Nystrom-attention encoder.
// Matrix math: V_WMMA_F32_16X16X4_F32 (wave32, exact fp32).
// B-operand staging: GLOBAL_LOAD_ASYNC_TO_LDS_B128 (ASYNCcnt), double-buffered
// so the DMA of chunk i+1 overlaps the WMMA consumption of chunk i.
// ---------------------------------------------------------------------------

typedef float v2f __attribute__((ext_vector_type(2)));
typedef float v8f __attribute__((ext_vector_type(8)));

__device__ __forceinline__ v8f wmma_f32(v2f a, v2f b, v8f c) {
  // D(16x16) = A(16x4) @ B(4x16) + C
  return __builtin_amdgcn_wmma_f32_16x16x4_f32(false, a, false, b, (short)0, c, false, false);
}

// ---------------------------------------------------------------------------
// WG-cooperative fp32 WMMA GEMM (B row-major):  C = alpha * A @ B (+bias)(+res)
//   One 256-thread workgroup computes a 256x64 C tile (8 waves x 32x64).
//   Per 64-K chunk, the 64x64 B block is async-DMA'd into LDS once per WG
//   (4 b128 issues/thread), ping-pong buffered against the WMMA consume loop.
//   Requires M%256==0, N%64==0, K%64==0 (true for every call here).
// ---------------------------------------------------------------------------
__global__ __launch_bounds__(256) void gemm_f32_lds(
    const float* __restrict__ A, int lda, int sA,
    const float* __restrict__ B, int ldb, int sB,
    float* __restrict__ C, int ldc, int sC,
    int M, int N, int K, float alpha,
    const float* __restrict__ bias, const float* __restrict__ resid,
    int qScaleCols, float qScale)
{
  __shared__ __align__(16) float Bs[2][64 * 68];  // 2 x (64 K-rows x 64 cols, stride 68)
  const int tid  = threadIdx.x;
  const int lane = tid & 31;
  const int wv   = tid >> 5;                    // wave id 0..7
  const int r    = lane & 15;
  const int hi   = lane >> 4;                   // K sub-pair selector
  const int nt64 = N >> 6;
  const int nt   = blockIdx.x % nt64;
  const int mb   = blockIdx.x / nt64;
  const int batch = blockIdx.y;

  const float* __restrict__ Ab = A + batch * sA;
  const float* __restrict__ Bb = B + batch * sB;
  float* __restrict__ Cb = C + batch * sC;

  const int row0  = mb * 256 + wv * 32;         // this wave's 32 rows
  const int ncol0 = nt * 64;
  const float* a0p = Ab + (row0 + r) * lda + 2 * hi;        // rows row0+r
  const float* a1p = a0p + 16 * lda;                        // rows row0+16+r

  const uint32_t ldsBase = (uint32_t)(size_t)&Bs[0][0];
  // per-thread staging pattern: id = it*256+tid -> K-row (id>>4), 16B col group (id&15)
  const int kr0 = tid >> 4;                     // K row for it=0 (0..15), +16 per it
  const int cg  = tid & 15;                     // 16B col group 0..15
  const uint32_t ldsThr = (uint32_t)(kr0 * 68 + cg * 4) * 4u;
  const int gThr = kr0 * ldb + ncol0 + cg * 4;

#define ISSUE_CHUNK(k0, buf)                                                     \
  {                                                                              \
    _Pragma("unroll")                                                            \
    for (int it = 0; it < 4; ++it) {                                             \
      const uint32_t ldsAddr = ldsBase + (uint32_t)(buf) * 17408u +              \
                               ldsThr + (uint32_t)(it * 16 * 68 * 4);            \
      const uint32_t goff = (uint32_t)((gThr + ((k0) + it * 16) * ldb) * 4);     \
      asm volatile("global_load_async_to_lds_b128 %0, %1, %2"                    \
                   :: "v"(ldsAddr), "v"(goff), "s"(Bb) : "memory");              \
    }                                                                            \
  }

  v8f acc[8] = {};

  ISSUE_CHUNK(0, 0);
  for (int k0 = 0; k0 < K; k0 += 64) {
    const int cur = (k0 >> 6) & 1;
    const bool more = (k0 + 64) < K;
    if (more) {
      ISSUE_CHUNK(k0 + 64, cur ^ 1);
      // my 4 just-issued may remain in flight; chunk-k0 loads (older) are done
      asm volatile("s_wait_asynccnt 4" ::: "memory");
    } else {
      asm volatile("s_wait_asynccnt 0" ::: "memory");
    }
    __syncthreads();                            // all waves' chunk-k0 data in LDS
    const float* __restrict__ Bc = &Bs[cur][0];

#pragma unroll 4
    for (int kk4 = 0; kk4 < 64; kk4 += 4) {
      const int kk = kk4 + 2 * hi;
      const v2f a0 = *(const v2f*)(a0p + k0 + kk4);
      const v2f a1 = *(const v2f*)(a1p + k0 + kk4);
#pragma unroll
      for (int j = 0; j < 4; ++j) {
        const int col = j * 16 + r;
        v2f b;
        b.x = Bc[kk * 68 + col];
        b.y = Bc[(kk + 1) * 68 + col];
        acc[j]     = wmma_f32(a0, b, acc[j]);
        acc[4 + j] = wmma_f32(a1, b, acc[4 + j]);
      }
    }
    __syncthreads();                            // buffer consumed by all waves
  }
#undef ISSUE_CHUNK

#pragma unroll
  for (int g = 0; g < 2; ++g)
#pragma unroll
    for (int j = 0; j < 4; ++j) {
      const int col = ncol0 + j * 16 + r;
      const float extra = (col < qScaleCols) ? qScale : 1.0f;
      const float bs = bias ? bias[col] : 0.0f;
#pragma unroll
      for (int e = 0; e < 8; ++e) {
        const int row = row0 + g * 16 + e + 8 * hi;   // D layout
        float v = acc[g * 4 + j][e] * alpha * extra + bs;
        if (resid) v += resid[row * ldc + col];
        Cb[row * ldc + col] = v;
      }
    }
}

// ---------------------------------------------------------------------------
// Per-wave fp32 WMMA GEMM with B stored (N x K) row-major (A @ B^T).
// Used only for the K=64 score GEMMs. Each wave: 16x64 tile, 4 accumulators.
// ---------------------------------------------------------------------------
__global__ __launch_bounds__(256) void gemm_f32_bt(
    const float* __restrict__ A, int lda, int sA,
    const float* __restrict__ B, int ldb, int sB,
    float* __restrict__ C, int ldc, int sC,
    int M, int N, int K)
{
  const int lane = threadIdx.x;
  const int r    = lane & 15;
  const int hi   = lane >> 4;
  const int nt64 = N >> 6;
  const int mt16 = M >> 4;
  const int waveId = blockIdx.x * 8 + threadIdx.y;
  if (waveId >= mt16 * nt64) return;     // wave-uniform: EXEC stays all-ones
  const int mt = waveId / nt64;
  const int nt = waveId % nt64;
  const int batch = blockIdx.y;

  const float* __restrict__ Ab = A + batch * sA;
  const float* __restrict__ Bb = B + batch * sB;
  float* __restrict__ Cb = C + batch * sC;

  v8f acc[4] = {};
  const int arow  = (mt * 16 + r) * lda;
  const int nbase = nt * 64 + r;
  const int koff  = 2 * hi;

  for (int k0 = 0; k0 < K; k0 += 8) {
    const int kk = k0 + koff;
    v2f a0 = *(const v2f*)(Ab + arow + kk);
    v2f a1 = *(const v2f*)(Ab + arow + kk + 4);
    v2f b0[4], b1[4];
#pragma unroll
    for (int j = 0; j < 4; ++j) {
      const int o = (nbase + j * 16) * ldb + kk;
      b0[j] = *(const v2f*)(Bb + o);
      b1[j] = *(const v2f*)(Bb + o + 4);
    }
#pragma unroll
    for (int j = 0; j < 4; ++j) acc[j] = wmma_f32(a0, b0[j], acc[j]);
#pragma unroll
    for (int j = 0; j < 4; ++j) acc[j] = wmma_f32(a1, b1[j], acc[j]);
  }

#pragma unroll
  for (int j = 0; j < 4; ++j) {
    const int col = nt * 64 + j * 16 + r;
#pragma unroll
    for (int e = 0; e < 8; ++e) {
      const int row = mt * 16 + e + 8 * hi;
      Cb[row * ldc + col] = acc[j][e];
    }
  }
}

// ---------------------------------------------------------------------------
// Landmark means: out[h][jm][c] = mean_{t in [jm*32,(jm+1)*32)} qkv[t][colOff+h*64+c]
// ---------------------------------------------------------------------------
__global__ void landmark_mean(const float* __restrict__ qkv, float* __restrict__ out, int colOff) {
  const int idx = blockIdx.x * 256 + threadIdx.x;     // 8*256*64 = 131072
  const int hh = idx >> 14;
  const int jm = (idx >> 6) & 255;
  const int c  = idx & 63;
  const int base = jm * 32 * 1536 + colOff + hh * 64 + c;
  float s = 0.0f;
#pragma unroll 8
  for (int t = 0; t < 32; ++t) s += qkv[base + t * 1536];
  out[idx] = s * (1.0f / 32.0f);
}

// ---------------------------------------------------------------------------
// Row softmax: one 256-thread block per row; src/dst may alias.
// ---------------------------------------------------------------------------
__global__ void softmax_rows(const float* __restrict__ src, float* __restrict__ dst,
                             int cols, int ld) {
  __shared__ float red[256];
  const int row = blockIdx.x;
  const float* s = src + row * ld;
  float* d = dst + row * ld;
  float m = -1e30f;
  for (int j = threadIdx.x; j < cols; j += 256) m = fmaxf(m, s[j]);
  red[threadIdx.x] = m; __syncthreads();
  for (int o = 128; o > 0; o >>= 1) {
    if (threadIdx.x < o) red[threadIdx.x] = fmaxf(red[threadIdx.x], red[threadIdx.x + o]);
    __syncthreads();
  }
  m = red[0]; __syncthreads();
  float sum = 0.0f;
  for (int j = threadIdx.x; j < cols; j += 256) sum += expf(s[j] - m);
  red[threadIdx.x] = sum; __syncthreads();
  for (int o = 128; o > 0; o >>= 1) {
    if (threadIdx.x < o) red[threadIdx.x] += red[threadIdx.x + o];
    __syncthreads();
  }
  const float inv = 1.0f / red[0];
  for (int j = threadIdx.x; j < cols; j += 256) d[j] = expf(s[j] - m) * inv;
}

// ---------------------------------------------------------------------------
// pinv helpers
// ---------------------------------------------------------------------------
__global__ void abs_sums(const float* __restrict__ x, float* __restrict__ rowsum,
                         float* __restrict__ colsum) {
  int idx = blockIdx.x * 256 + threadIdx.x;           // 4096 threads
  if (idx < 2048) {
    const int h = idx >> 8, i = idx & 255;
    const int base = h * 65536 + i * 256;
    float s = 0.0f;
    for (int j = 0; j < 256; ++j) s += fabsf(x[base + j]);
    rowsum[idx] = s;
  } else {
    idx -= 2048;
    const int h = idx >> 8, j = idx & 255;
    const int base = h * 65536 + j;
    float s = 0.0f;
    for (int i = 0; i < 256; ++i) s += fabsf(x[base + i * 256]);
    colsum[idx] = s;
  }
}

__global__ void pinv_scale(const float* __restrict__ rs, const float* __restrict__ cs,
                           float* __restrict__ inv) {
  __shared__ float a[256], b[256];
  float m1 = -1e30f, m2 = -1e30f;
  for (int i = threadIdx.x; i < 2048; i += 256) { m1 = fmaxf(m1, rs[i]); m2 = fmaxf(m2, cs[i]); }
  a[threadIdx.x] = m1; b[threadIdx.x] = m2; __syncthreads();
  for (int o = 128; o > 0; o >>= 1) {
    if (threadIdx.x < o) {
      a[threadIdx.x] = fmaxf(a[threadIdx.x], a[threadIdx.x + o]);
      b[threadIdx.x] = fmaxf(b[threadIdx.x], b[threadIdx.x + o]);
    }
    __syncthreads();
  }
  if (threadIdx.x == 0) inv[0] = 1.0f / (a[0] * b[0]);
}

__global__ void transpose_scale(const float* __restrict__ x, float* __restrict__ z,
                                const float* __restrict__ invscale) {
  const int idx = blockIdx.x * 256 + threadIdx.x;     // 524288
  const int h = idx >> 16, rem = idx & 65535;
  const int i = rem >> 8, j = rem & 255;
  z[idx] = x[h * 65536 + j * 256 + i] * invscale[0];
}

__global__ void diag_sub(const float* __restrict__ s, float* __restrict__ t, float c) {
  const int idx = blockIdx.x * 256 + threadIdx.x;     // 524288
  const int rem = idx & 65535;
  const int i = rem >> 8, j = rem & 255;
  t[idx] = ((i == j) ? c : 0.0f) - s[idx];
}

// ---------------------------------------------------------------------------
// Fused attn1 stage: per wave, 16 rows of one head.
//   S(16x256) = q_rows @ k_l^T  (WMMA)  -> LDS
//   row softmax over 256 (in LDS, wave32 shuffle pairing)
//   O(16x64) = S @ w_mid                 (WMMA, A-fragments from LDS)
// blockDim = (32,2); grid = 8 heads * 512 row-blocks / 2.
// ---------------------------------------------------------------------------
__global__ __launch_bounds__(64) void attn1_fused(
    const float* __restrict__ qkv, const float* __restrict__ kl,
    const float* __restrict__ wmid, float* __restrict__ out_nh)
{
  __shared__ float S[2][16][260];                      // +4 pad vs bank conflicts
  const int lane = threadIdx.x, w = threadIdx.y;
  const int r = lane & 15, hi = lane >> 4;
  const int gid = blockIdx.x * 2 + w;
  const int hh = gid >> 9;
  const int i0 = (gid & 511) * 16;
  const int qbase = i0 * 1536 + hh * 64 + r * 1536;    // q row for this lane
  const int klb   = hh * 16384;

  // Phase 1: scores -> LDS
  for (int nt = 0; nt < 16; ++nt) {
    v8f acc = {};
    const int colb = klb + (nt * 16 + r) * 64;
#pragma unroll 2
    for (int k0 = 0; k0 < 64; k0 += 4) {
      const int kk = k0 + 2 * hi;
      v2f a = *(const v2f*)(qkv + qbase + kk);
      v2f b = *(const v2f*)(kl + colb + kk);
      acc = wmma_f32(a, b, acc);
    }
#pragma unroll
    for (int e = 0; e < 8; ++e) S[w][e + 8 * hi][nt * 16 + r] = acc[e];
  }

  // Phase 2: row softmax (2 lanes per row; pair-combine via shuffle)
  {
    const int row = lane >> 1, part = lane & 1;
    float* sr = &S[w][row][0];
    float m = -1e30f;
    for (int j = part * 128; j < part * 128 + 128; ++j) m = fmaxf(m, sr[j]);
    m = fmaxf(m, __shfl_xor(m, 1, 32));
    float sum = 0.0f;
    for (int j = part * 128; j < part * 128 + 128; ++j) sum += expf(sr[j] - m);
    sum += __shfl_xor(sum, 1, 32);
    const float inv = 1.0f / sum;
    for (int j = part * 128; j < part * 128 + 128; ++j) sr[j] = expf(sr[j] - m) * inv;
  }

  // Phase 3: O = P @ w_mid  (A from LDS in 16x4 fragment layout)
  v8f acc[4] = {};
  const int wmb = hh * 16384;
  for (int k0 = 0; k0 < 256; k0 += 4) {
    const int kk = k0 + 2 * hi;
    v2f a; a.x = S[w][r][kk]; a.y = S[w][r][kk + 1];
    const int o = wmb + kk * 64 + r;
#pragma unroll
    for (int j = 0; j < 4; ++j) {
      v2f b; b.x = wmid[o + j * 16]; b.y = wmid[o + j * 16 + 64];
      acc[j] = wmma_f32(a, b, acc[j]);
    }
  }

  // Phase 4: store (n, h*dh) layout
#pragma unroll
  for (int j = 0; j < 4; ++j)
#pragma unroll
    for (int e = 0; e < 8; ++e)
      out_nh[(i0 + e + 8 * hi) * 512 + hh * 64 + j * 16 + r] = acc[j][e];
}

// ---------------------------------------------------------------------------
// Depthwise conv residual (kernel 33 over sequence), added into out_nh.
// ---------------------------------------------------------------------------
__global__ void conv_add(const float* __restrict__ qkv, const float* __restrict__ cw,
                         float* __restrict__ out_nh) {
  const int idx = blockIdx.x * 256 + threadIdx.x;     // 8192*512
  const int i = idx >> 9;
  const int c512 = idx & 511;
  const int hh = c512 >> 6;
  float acc = 0.0f;
#pragma unroll
  for (int k = 0; k < 33; ++k) {
    const int t = i + k - 16;
    if (t >= 0 && t < 8192) acc += cw[hh * 33 + k] * qkv[t * 1536 + 1024 + c512];
  }
  out_nh[idx] += acc;
}

// ---------------------------------------------------------------------------
// Sparse neighbor aggregation: A_raw[r] += vals[e] * (q[r].k[c]) / 16
// ---------------------------------------------------------------------------
__global__ void zero_f32(float* __restrict__ p, int n) {
  const int i = blockIdx.x * 256 + threadIdx.x;
  if (i < n) p[i] = 0.0f;
}

__global__ void sparse_scores(const int* __restrict__ rows, const int* __restrict__ cols,
                              const float* __restrict__ vals,
                              const float* __restrict__ qp, const float* __restrict__ kp,
                              float* __restrict__ Araw, int E) {
  const int e = blockIdx.x * 256 + threadIdx.x;
  if (e >= E) return;
  const int rr = rows[e], cc = cols[e];
  const float4* __restrict__ qa = (const float4*)(qp + rr * 256);
  const float4* __restrict__ ka = (const float4*)(kp + cc * 256);
  float s = 0.0f;
#pragma unroll 8
  for (int t = 0; t < 64; ++t) {
    const float4 a = qa[t], b = ka[t];
    s += a.x * b.x + a.y * b.y + a.z * b.z + a.w * b.w;
  }
  atomicAdd(&Araw[rr], vals[e] * s * 0.0625f);
}

// ---------------------------------------------------------------------------
// Gated fusion: xo = xl*2*sw + 2*enc*(1-sw); xl=alpha[i]*value; wei=sigmoid(-xl)
// ---------------------------------------------------------------------------
__global__ void fuse(const float* __restrict__ value, const float* __restrict__ enc,
                     const float* __restrict__ alpha, float* __restrict__ xo) {
  const int idx = blockIdx.x * 256 + threadIdx.x;     // 8388608
  const int i = idx >> 10;
  const float xl = alpha[i] * value[idx];
  const float wei = 1.0f / (1.0f + expf(xl));         // sigmoid(-xl)
  const float sw = wei * wei;
  xo[idx] = xl * 2.0f * sw + 2.0f * enc[idx] * (1.0f - sw);
}

// ---------------------------------------------------------------------------
// Host orchestration
// ---------------------------------------------------------------------------
static inline void launch_gemm_lds(hipStream_t st,
                                   const float* A, int lda, int sA,
                                   const float* B, int ldb, int sB,
                                   float* C, int ldc, int sC,
                                   int M, int N, int K, int batch,
                                   float alpha = 1.0f,
                                   const float* bias = nullptr, const float* resid = nullptr,
                                   int qCols = 0, float qScale = 1.0f) {
  dim3 grid((M / 256) * (N / 64), batch);
  gemm_f32_lds<<<grid, 256, 0, st>>>(A, lda, sA, B, ldb, sB, C, ldc, sC,
                                     M, N, K, alpha, bias, resid, qCols, qScale);
}

static inline void launch_gemm_bt(hipStream_t st,
                                  const float* A, int lda, int sA,
                                  const float* B, int ldb, int sB,
                                  float* C, int ldc, int sC,
                                  int M, int N, int K, int batch) {
  const int tiles = (M / 16) * (N / 64);
  dim3 grid((tiles + 7) / 8, batch), blk(32, 8);
  gemm_f32_bt<<<grid, blk, 0, st>>>(A, lda, sA, B, ldb, sB, C, ldc, sC, M, N, K);
}

extern "C" void kernel_launch(void* const* d_in, const int* in_sizes, int n_in,
                              void* d_out, int out_size, void* d_ws, size_t ws_size,
                              hipStream_t stream) {
  const float* dense    = (const float*)d_in[0];
  const int*   adj_rows = (const int*)d_in[1];
  const int*   adj_cols = (const int*)d_in[2];
  const float* adj_vals = (const float*)d_in[3];
  const float* wq       = (const float*)d_in[4];
  const float* wk       = (const float*)d_in[5];
  const float* wv_w     = (const float*)d_in[6];
  const float* wv_b     = (const float*)d_in[7];
  const float* qkv_w    = (const float*)d_in[8];
  const float* out_w    = (const float*)d_in[9];
  const float* out_b    = (const float*)d_in[10];
  const float* conv_w   = (const float*)d_in[11];
  const int E = in_sizes[1];

  // Workspace layout (floats); enc/value alias attn3, qp/kp alias out_nh.
  float* W = (float*)d_ws;
  float* qkv    = W; W += 12582912;   // 8192 x 1536
  float* ql     = W; W += 131072;     // 8 x 256 x 64
  float* kl     = W; W += 131072;
  float* attn2  = W; W += 524288;     // 8 x 256 x 256
  float* z1     = W; W += 524288;
  float* z2     = W; W += 524288;
  float* xz     = W; W += 524288;
  float* tmp    = W; W += 524288;
  float* rowsum = W; W += 2048;
  float* colsum = W; W += 2048;
  float* invsc  = W; W += 32;
  float* alphaB = W; W += 8192;
  float* attn3  = W; W += 16777216;   // 8 x 256 x 8192
  float* attn3v = W; W += 131072;
  float* wmid   = W; W += 131072;
  float* out_nh = W; W += 4194304;    // 8192 x 512
  float* enc    = attn3;              // alias (attn3 dead after attn3v)
  float* value  = attn3 + 8388608;    // alias
  float* qp     = out_nh;             // alias (out_nh dead after enc gemm)
  float* kp     = out_nh + 2097152;   // alias

  float* xo   = (float*)d_out;
  float* Araw = (float*)d_out + 8388608;

  // 1. qkv = dense @ qkv_w ; q columns (<512) pre-scaled by dh^-0.5 = 1/8
  launch_gemm_lds(stream, dense, 1024, 0, qkv_w, 1536, 0, qkv, 1536, 0,
                  8192, 1536, 1024, 1, 1.0f, nullptr, nullptr, 512, 0.125f);

  // 2. landmark means (q already scaled)
  landmark_mean<<<512, 256, 0, stream>>>(qkv, ql, 0);
  landmark_mean<<<512, 256, 0, stream>>>(qkv, kl, 512);

  // 3. attn2 = softmax(q_l @ k_l^T) per head
  launch_gemm_bt(stream, ql, 64, 16384, kl, 64, 16384, attn2, 256, 65536,
                 256, 256, 64, 8);
  softmax_rows<<<2048, 256, 0, stream>>>(attn2, attn2, 256, 256);

  // 4. Moore-Penrose pinv (6 Newton-Schulz iterations, batched over heads)
  abs_sums<<<16, 256, 0, stream>>>(attn2, rowsum, colsum);
  pinv_scale<<<1, 256, 0, stream>>>(rowsum, colsum, invsc);
  transpose_scale<<<2048, 256, 0, stream>>>(attn2, z1, invsc);
  float* zc = z1; float* zn = z2;
  for (int it = 0; it < 6; ++it) {
    launch_gemm_lds(stream, attn2, 256, 65536, zc, 256, 65536, xz, 256, 65536, 256, 256, 256, 8);
    diag_sub<<<2048, 256, 0, stream>>>(xz, tmp, 7.0f);
    launch_gemm_lds(stream, xz, 256, 65536, tmp, 256, 65536, zn, 256, 65536, 256, 256, 256, 8);
    diag_sub<<<2048, 256, 0, stream>>>(zn, tmp, 15.0f);
    launch_gemm_lds(stream, xz, 256, 65536, tmp, 256, 65536, zn, 256, 65536, 256, 256, 256, 8);
    diag_sub<<<2048, 256, 0, stream>>>(zn, tmp, 13.0f);
    launch_gemm_lds(stream, zc, 256, 65536, tmp, 256, 65536, zn, 256, 65536, 256, 256, 256, 8, 0.25f);
    float* t = zc; zc = zn; zn = t;
  }

  // 5. attn3 = softmax(q_l @ k^T) ; attn3v = attn3 @ v
  launch_gemm_bt(stream, ql, 64, 16384, qkv + 512, 1536, 64, attn3, 8192, 2097152,
                 256, 8192, 64, 8);
  softmax_rows<<<2048, 256, 0, stream>>>(attn3, attn3, 8192, 8192);
  launch_gemm_lds(stream, attn3, 8192, 2097152, qkv + 1024, 1536, 64, attn3v, 64, 16384,
                  256, 64, 8192, 8);

  // 6. w_mid = pinv(attn2) @ attn3v
  launch_gemm_lds(stream, zc, 256, 65536, attn3v, 64, 16384, wmid, 64, 16384,
                  256, 64, 256, 8);

  // 7. fused attn1: out_nh = softmax(q @ k_l^T) @ w_mid
  attn1_fused<<<2048, dim3(32, 2), 0, stream>>>(qkv, kl, wmid, out_nh);

  // 8. + depthwise conv residual on v
  conv_add<<<16384, 256, 0, stream>>>(qkv, conv_w, out_nh);

  // 9. enc = out_nh @ out_w + out_b + dense
  launch_gemm_lds(stream, out_nh, 512, 0, out_w, 1024, 0, enc, 1024, 0,
                  8192, 1024, 512, 1, 1.0f, out_b, dense);

  // 10. q/k projections
  launch_gemm_lds(stream, enc, 1024, 0, wq, 256, 0, qp, 256, 0, 8192, 256, 1024, 1);
  launch_gemm_lds(stream, enc, 1024, 0, wk, 256, 0, kp, 256, 0, 8192, 256, 1024, 1);

  // 11. sparse neighbor aggregation -> A_raw (second output, in d_out tail)
  zero_f32<<<32, 256, 0, stream>>>(Araw, 8192);
  sparse_scores<<<(E + 255) / 256, 256, 0, stream>>>(adj_rows, adj_cols, adj_vals,
                                                     qp, kp, Araw, E);
  softmax_rows<<<1, 256, 0, stream>>>(Araw, alphaB, 8192, 8192);

  // 12. value = dense @ wv_w + wv_b ; gated fusion -> xo
  launch_gemm_lds(stream, dense, 1024, 0, wv_w, 1024, 0, value, 1024, 0,
                  8192, 1024, 1024, 1, 1.0f, wv_b, nullptr);
  fuse<<<32768, 256, 0, stream>>>(value, enc, alphaB, xo);
}